// PagedTransformer_54649163874778
// MI455X (gfx1250) — compile-verified
//
#include <hip/hip_runtime.h>
#include <hip/hip_bf16.h>
#include <math.h>

// ---------------------------------------------------------------------------
// Problem constants (match reference)
// ---------------------------------------------------------------------------
#define BB   8       // batch
#define LL   2       // layers
#define DD   2048    // model dim
#define HH   16      // heads
#define HD   128     // head dim
#define DFF  8192    // ffn dim
#define BS   16      // kv block size
#define MB   128     // logical blocks per seq
#define TT   2048    // MB*BS tokens of context
#define SCALEF 0.08838834764831845f   // 1/sqrt(128)
#define EPSF   1e-5f

typedef __attribute__((ext_vector_type(2))) float v2f;
typedef __attribute__((ext_vector_type(8))) float v8f;

// ---------------------------------------------------------------------------
// RMSNorm:  out[b,:] = x[b,:] * rsqrt(mean(x^2)+eps) * w
// ---------------------------------------------------------------------------
__global__ __launch_bounds__(256) void rmsnorm_kernel(
    const float* __restrict__ x, const float* __restrict__ w,
    float* __restrict__ out, int Dn)
{
    __shared__ float red[256];
    const int b = blockIdx.x, tid = threadIdx.x;
    const float* xb = x + (size_t)b * Dn;
    float s = 0.f;
    for (int i = tid; i < Dn; i += 256) { float v = xb[i]; s += v * v; }
    red[tid] = s; __syncthreads();
    #pragma unroll
    for (int k = 128; k > 0; k >>= 1) {
        if (tid < k) red[tid] += red[tid + k];
        __syncthreads();
    }
    const float scale = rsqrtf(red[0] / (float)Dn + EPSF);
    for (int i = tid; i < Dn; i += 256)
        out[(size_t)b * Dn + i] = xb[i] * scale * w[i];
}

// ---------------------------------------------------------------------------
// Skinny GEMV via WMMA f32 16x16x4, K-split for wave-level parallelism:
//   part[(mat*SPLIT+kc), m, n] = sum_{k in slice kc} A[m,k] * W_mat[n,k]
// A: [8, K] row-major (rows 8..15 of the WMMA A-tile mirror rows 0..7 and are
// discarded).  W: [N, K] row-major.  One wave per (16-col tile, K-slice).
// grid = dim3(NMAT * N/16, SPLIT), block = 32.
// WMMA operand placement (wave32):
//   A 16x4:  lanes 0-15 rows M=0..15; vgpr0=K0/K2, vgpr1=K1/K3
//            -> per-lane float2 at khi = 2*(lane>>4)
//   B 4x16:  lanes 0-15 cols N=0..15, same K striping -> same float2 pattern
//   C 16x16: vgpr j = row M=j (lanes 0-15) / M=j+8 (lanes 16-31)
// ---------------------------------------------------------------------------
template<int K, int N, int SPLIT>
__global__ __launch_bounds__(32) void gemv_wmma_kernel(
    const float* __restrict__ A,
    const float* __restrict__ W0, const float* __restrict__ W1,
    const float* __restrict__ W2,
    float* __restrict__ part)
{
    constexpr int tilesPer = N / 16;
    constexpr int KS = K / SPLIT;          // K-slice length (multiple of 32)

    const int tile = blockIdx.x;
    const int mat  = tile / tilesPer;      // constant divisor -> cheap
    const int n0   = (tile % tilesPer) << 4;
    const int kc   = blockIdx.y;

    const float* W = (mat == 0) ? W0 : ((mat == 1) ? W1 : W2);

    const int lane = threadIdx.x;
    const int mrow = lane & 15;
    const int khi  = (lane >> 4) << 1;     // 0 or 2

    const float* Ap = A + (size_t)(mrow & 7) * K + kc * KS + khi;
    const float* Wp = W + (size_t)(n0 + mrow) * K + kc * KS + khi;

    v8f c;
    #pragma unroll
    for (int i = 0; i < 8; ++i) c[i] = 0.0f;

    for (int kb = 0; kb < KS; kb += 32) {
        // stay ~640B ahead on each streamed weight row
        __builtin_prefetch(Wp + kb + 160, 0, 0);
        #pragma unroll
        for (int kk = 0; kk < 32; kk += 4) {
            v2f a  = *(const v2f*)(Ap + kb + kk);
            v2f bb = *(const v2f*)(Wp + kb + kk);
            c = __builtin_amdgcn_wmma_f32_16x16x4_f32(
                    /*neg_a=*/false, a, /*neg_b=*/false, bb,
                    /*c_mod=*/(short)0, c, /*reuse_a=*/false, /*reuse_b=*/false);
        }
    }

    if (lane < 16) {
        const int n = n0 + lane;
        const size_t base = ((size_t)(mat * SPLIT + kc) * 8) * N;
        #pragma unroll
        for (int j = 0; j < 8; ++j)
            part[base + (size_t)j * N + n] = c[j];
    }
}

// ---------------------------------------------------------------------------
// Sum K-slices, add optional residual (mat 0 only), scatter to per-matrix outs.
// ---------------------------------------------------------------------------
template<int NMAT, int SPLIT>
__global__ __launch_bounds__(256) void reduce_kernel(
    const float* __restrict__ part,
    float* __restrict__ O0, float* __restrict__ O1, float* __restrict__ O2,
    const float* __restrict__ resid, int N)
{
    const int idx   = blockIdx.x * 256 + threadIdx.x;
    const int perM  = 8 * N;
    if (idx >= NMAT * perM) return;
    const int mat = idx / perM;
    const int rem = idx - mat * perM;          // m*N + n
    const float* p = part + (size_t)mat * SPLIT * perM + rem;
    float s = 0.f;
    #pragma unroll
    for (int k = 0; k < SPLIT; ++k) s += p[(size_t)k * perM];
    if (mat == 0 && resid) s += resid[rem];
    float* O = (mat == 0) ? O0 : ((mat == 1) ? O1 : O2);
    O[rem] = s;
}

// ---------------------------------------------------------------------------
// Paged attention for one (b, h). grid = B*H blocks, 256 threads (8 waves).
// Gathers K/V through the block table; substitutes this layer's fresh k/v
// where (physical block, offset) == slot_mapping (functional reshape_and_cache).
// ---------------------------------------------------------------------------
__global__ __launch_bounds__(256) void attn_kernel(
    const float* __restrict__ q_s, const float* __restrict__ k_s,
    const float* __restrict__ v_s,
    const float* __restrict__ key_heap, const float* __restrict__ val_heap,
    const int* __restrict__ bt,          // [B, MB] for this layer
    const int* __restrict__ slot_map,    // [B]     for this layer
    const int* __restrict__ ctx_lens,    // [B]
    float* __restrict__ o_s)             // [B, H*HD]
{
    __shared__ float s_lds[TT];
    __shared__ float red[256];
    __shared__ int   btab[MB];
    __shared__ __align__(16) float qv[HD];

    const int b = blockIdx.x >> 4;
    const int h = blockIdx.x & 15;
    const int tid = threadIdx.x;

    if (tid < MB) btab[tid] = bt[b * MB + tid];
    if (tid < HD) qv[tid]   = q_s[(size_t)b * DD + h * HD + tid];
    __syncthreads();

    const int slot  = slot_map[b];
    const int blk_s = slot >> 4, off_s = slot & 15;
    const int ctx   = ctx_lens[b];

    const int lane = tid & 31, wave = tid >> 5;
    const float4 q4 = *(const float4*)(&qv[lane * 4]);
    const float* kscr = k_s + (size_t)b * DD + h * HD + lane * 4;

    // ---- phase 1: scores s[t] = scale * q . K[t] -----------------------------
    for (int t = wave; t < TT; t += 8) {
        const int phys = btab[t >> 4];
        const int off  = t & 15;
        const float* kp = (phys == blk_s && off == off_s)
            ? kscr
            : key_heap + (((size_t)phys * HH + h) * BS + off) * HD + lane * 4;
        const float4 k4 = *(const float4*)kp;
        float d = q4.x * k4.x + q4.y * k4.y + q4.z * k4.z + q4.w * k4.w;
        #pragma unroll
        for (int s = 16; s > 0; s >>= 1) d += __shfl_xor(d, s, 32);
        if (lane == 0) {
            float sc = d * SCALEF;
            if (t >= ctx) sc = -3.0e38f;
            s_lds[t] = sc;
        }
    }
    __syncthreads();

    // ---- phase 2: softmax ----------------------------------------------------
    float m = -3.0e38f;
    for (int i = tid; i < TT; i += 256) m = fmaxf(m, s_lds[i]);
    red[tid] = m; __syncthreads();
    #pragma unroll
    for (int s = 128; s > 0; s >>= 1) {
        if (tid < s) red[tid] = fmaxf(red[tid], red[tid + s]);
        __syncthreads();
    }
    m = red[0];
    __syncthreads();

    float zs = 0.f;
    for (int i = tid; i < TT; i += 256) {
        const float e = __expf(s_lds[i] - m);
        s_lds[i] = e;
        zs += e;
    }
    red[tid] = zs; __syncthreads();
    #pragma unroll
    for (int s = 128; s > 0; s >>= 1) {
        if (tid < s) red[tid] += red[tid + s];
        __syncthreads();
    }
    const float Z = red[0];

    // ---- phase 3: o[d] = sum_t p[t] * V[t][d] --------------------------------
    const int d    = tid & 127;
    const int half = tid >> 7;
    const float* vscr = v_s + (size_t)b * DD + h * HD;
    float acc = 0.f;
    for (int t = half * 1024; t < half * 1024 + 1024; ++t) {
        const int phys = btab[t >> 4];
        const int off  = t & 15;
        const float vv = (phys == blk_s && off == off_s)
            ? vscr[d]
            : val_heap[(((size_t)phys * HH + h) * BS + off) * HD + d];
        acc += s_lds[t] * vv;
    }
    __syncthreads();        // all threads have read red[0] before reuse
    red[tid] = acc;
    __syncthreads();
    if (tid < 128)
        o_s[(size_t)b * DD + h * HD + d] = (red[tid] + red[tid + 128]) / Z;
}

// ---------------------------------------------------------------------------
// SwiGLU gate: out[i] = silu(g[i]) * u[i]
// ---------------------------------------------------------------------------
__global__ __launch_bounds__(256) void silumul_kernel(
    const float* __restrict__ g, const float* __restrict__ u,
    float* __restrict__ out, int n)
{
    const int i = blockIdx.x * 256 + threadIdx.x;
    if (i < n) {
        const float x = g[i];
        out[i] = (x / (1.f + __expf(-x))) * u[i];
    }
}

// ---------------------------------------------------------------------------
// Host-side pipeline
// ---------------------------------------------------------------------------
extern "C" void kernel_launch(void* const* d_in, const int* in_sizes, int n_in,
                              void* d_out, int out_size, void* d_ws, size_t ws_size,
                              hipStream_t stream)
{
    const float* x_in      = (const float*)d_in[0];
    const float* key_heap  = (const float*)d_in[1];
    const float* val_heap  = (const float*)d_in[2];
    const int*   btables   = (const int*)  d_in[3];
    const int*   slot_map  = (const int*)  d_in[4];
    const int*   ctx_lens  = (const int*)  d_in[5];
    const float* wq        = (const float*)d_in[6];
    const float* wk        = (const float*)d_in[7];
    const float* wv        = (const float*)d_in[8];
    const float* wo        = (const float*)d_in[9];
    const float* w1        = (const float*)d_in[10];
    const float* w2        = (const float*)d_in[11];
    const float* w3        = (const float*)d_in[12];
    const float* norm1_w   = (const float*)d_in[13];
    const float* norm2_w   = (const float*)d_in[14];
    const float* final_w   = (const float*)d_in[15];

    float* ws   = (float*)d_ws;
    float* h    = ws;                 // [8, 2048]
    float* q    = ws + 16384;         // [8, 2048]
    float* kk   = ws + 32768;         // [8, 2048]
    float* vv   = ws + 49152;         // [8, 2048]
    float* o    = ws + 65536;         // [8, 2048]
    float* xa   = ws + 81920;         // [8, 2048] residual stream (current)
    float* xb   = ws + 98304;         // [8, 2048] residual stream (post-attn)
    float* g    = ws + 114688;        // [8, 8192]
    float* u    = ws + 180224;        // [8, 8192]
    float* part = ws + 245760;        // up to [2*4][8, 8192] = 524288 floats

    hipMemcpyAsync(xa, x_in, (size_t)BB * DD * sizeof(float),
                   hipMemcpyDeviceToDevice, stream);

    const size_t WSZ = (size_t)DD * DD;    // attn weight matrix size
    const size_t FSZ = (size_t)DFF * DD;   // ffn weight matrix size

    for (int l = 0; l < LL; ++l) {
        // -- attention block --------------------------------------------------
        rmsnorm_kernel<<<BB, 256, 0, stream>>>(xa, norm1_w + l * DD, h, DD);

        gemv_wmma_kernel<DD, DD, 4><<<dim3(3 * (DD / 16), 4), 32, 0, stream>>>(
            h, wq + l * WSZ, wk + l * WSZ, wv + l * WSZ, part);
        reduce_kernel<3, 4><<<(3 * 8 * DD + 255) / 256, 256, 0, stream>>>(
            part, q, kk, vv, nullptr, DD);

        attn_kernel<<<BB * HH, 256, 0, stream>>>(
            q, kk, vv, key_heap, val_heap,
            btables + l * BB * MB, slot_map + l * BB, ctx_lens, o);

        gemv_wmma_kernel<DD, DD, 8><<<dim3(DD / 16, 8), 32, 0, stream>>>(
            o, wo + l * WSZ, nullptr, nullptr, part);
        reduce_kernel<1, 8><<<(8 * DD + 255) / 256, 256, 0, stream>>>(
            part, xb, nullptr, nullptr, /*resid=*/xa, DD);

        // -- FFN block --------------------------------------------------------
        rmsnorm_kernel<<<BB, 256, 0, stream>>>(xb, norm2_w + l * DD, h, DD);

        gemv_wmma_kernel<DD, DFF, 4><<<dim3(2 * (DFF / 16), 4), 32, 0, stream>>>(
            h, w1 + l * FSZ, w3 + l * FSZ, nullptr, part);
        reduce_kernel<2, 4><<<(2 * 8 * DFF + 255) / 256, 256, 0, stream>>>(
            part, g, u, nullptr, nullptr, DFF);

        silumul_kernel<<<(BB * DFF + 255) / 256, 256, 0, stream>>>(
            g, u, g, BB * DFF);

        gemv_wmma_kernel<DFF, DD, 8><<<dim3(DD / 16, 8), 32, 0, stream>>>(
            g, w2 + l * FSZ, nullptr, nullptr, part);
        reduce_kernel<1, 8><<<(8 * DD + 255) / 256, 256, 0, stream>>>(
            part, xa, nullptr, nullptr, /*resid=*/xb, DD);
        // current residual stream is back in xa
    }

    rmsnorm_kernel<<<BB, 256, 0, stream>>>(xa, final_w, (float*)d_out, DD);
}